// trq_2b_46789373723186
// MI455X (gfx1250) — compile-verified
//
#include <hip/hip_runtime.h>

#define EPS2 1e-12f
#define RSQRT_IR 0.31622776601683794f  /* 1/sqrt(10) */

typedef __attribute__((ext_vector_type(2))) float v2f;
typedef __attribute__((ext_vector_type(8))) float v8f;

#if __has_builtin(__builtin_amdgcn_tanhf)
#define FAST_TANH(x) __builtin_amdgcn_tanhf(x)
#else
#define FAST_TANH(x) tanhf(x)
#endif

__device__ __forceinline__ float fast_sqrt(float x) { return __builtin_amdgcn_sqrtf(x); }
__device__ __forceinline__ float fast_rcp(float x) { return __builtin_amdgcn_rcpf(x); }

// One block per batch row b; 256 threads = 256 points = 8 waves = 16 WMMA tiles.
__global__ __launch_bounds__(256) void trq_2b_kernel(
    const float* __restrict__ v1b,   // (B,3)
    const float* __restrict__ invar, // (B,256,2)
    const float* __restrict__ v2b,   // (B,256,6)
    const float* __restrict__ W1,    // (2,10)
    const float* __restrict__ b1,    // (10)
    const float* __restrict__ W2,    // (10,10)
    const float* __restrict__ b2,    // (10)
    const float* __restrict__ lw1,   // (1,10)
    const float* __restrict__ nb1,   // (10)
    const float* __restrict__ lw2,   // (10,10)
    const float* __restrict__ nb2,   // (10)
    const float* __restrict__ wf,    // (10)
    float* __restrict__ out)         // (B,3)
{
  // Pre-padded weight tiles: unconditional ds_loads in the hot path (no exec-mask
  // predicated loads). sW1p is the 4x16 zero-padded W1, sW2p the 12x16 zero-padded W2.
  __shared__ float sW1p[4 * 16];
  __shared__ float sW2p[12 * 16];
  __shared__ float sB1[16];
  __shared__ float sB2[16];
  __shared__ float slw1[10], snb1[10], slw2[100], snb2[10], swf[10];
  __shared__ float ytile[8][2][16 * 20];  // per-wave, per-tile 16x16 (stride 20: conflict-free)
  __shared__ float red[8][4];

  const int t = threadIdx.x;
  const int b = blockIdx.x;
  const int lane = t & 31;
  const int wave = t >> 5;
  const int halfIdx = lane >> 4;  // 0: lanes 0-15, 1: lanes 16-31
  const int lm = lane & 15;

  // ---- stage small weights into LDS (zero-padded tiles) ----
  {
    const int k = t >> 4, n = t & 15;          // tile coords for thread t
    if (t < 64)  sW1p[t] = (k < 2  && n < 10) ? W1[k * 10 + n] : 0.f;
    if (t < 192) sW2p[t] = (k < 10 && n < 10) ? W2[k * 10 + n] : 0.f;
  }
  if (t < 100) slw2[t] = lw2[t];
  if (t < 16)  { sB1[t] = (t < 10) ? b1[t] : 0.f; sB2[t] = (t < 10) ? b2[t] : 0.f; }
  if (t < 10)  { slw1[t] = lw1[t]; snb1[t] = nb1[t]; snb2[t] = nb2[t]; swf[t] = wf[t]; }
  __syncthreads();

  // ---- per-point loads ----
  const size_t idx = (size_t)b * 256 + (size_t)t;
  const v2f inv = *(const v2f*)(invar + idx * 2);          // 8B aligned
  const float* vp = v2b + idx * 6 + 3;                     // only 4B aligned -> 3 scalar loads
  const float v0 = vp[0], v1 = vp[1], v2 = vp[2];

  // ================= y-chain via V_WMMA_F32_16X16X4_F32 =================
  // A 16x4 f32 layout: lanes 0-15 = M rows (VGPR0=K0, VGPR1=K1); lanes 16-31 = K2/K3 (zero pad).
  const float i0s = __shfl(inv.x, (lane + 16) & 31, 32);
  const float i1s = __shfl(inv.y, (lane + 16) & 31, 32);
  v2f a0, a1;
  a0.x = (lane < 16) ? inv.x : 0.f;  a0.y = (lane < 16) ? inv.y : 0.f;   // tile0: points wave*32+0..15
  a1.x = (lane < 16) ? i0s  : 0.f;   a1.y = (lane < 16) ? i1s  : 0.f;    // tile1: points wave*32+16..31

  // B = W1 padded 4x16 (B 4x16: VGPR0 = K row (0 or 2); VGPR1 = K row (1 or 3)); rows 2,3 are zero.
  v2f bw1;
  bw1.x = sW1p[(0 + 2 * halfIdx) * 16 + lm];
  bw1.y = sW1p[(1 + 2 * halfIdx) * 16 + lm];

  // C = bias broadcast over rows (C/D: VGPR r -> M=r+8*halfIdx, N=lm)
  const float cb1 = sB1[lm];
  v8f d0 = {cb1, cb1, cb1, cb1, cb1, cb1, cb1, cb1};
  v8f d1 = d0;

  d0 = __builtin_amdgcn_wmma_f32_16x16x4_f32(false, a0, false, bw1, (short)0, d0, false, false);
  d1 = __builtin_amdgcn_wmma_f32_16x16x4_f32(false, a1, false, bw1, (short)0, d1, false, false);

#pragma unroll
  for (int r = 0; r < 8; ++r) { d0[r] = fmaxf(d0[r], 0.f); d1[r] = fmaxf(d1[r], 0.f); }

  // Transpose Y1 (C layout -> A layout) via LDS
#pragma unroll
  for (int r = 0; r < 8; ++r) {
    const int m = r + 8 * halfIdx;
    ytile[wave][0][m * 20 + lm] = d0[r];
    ytile[wave][1][m * 20 + lm] = d1[r];
  }
  __syncthreads();

  // y2 = relu(Y1 @ W2 + b2): K=10 padded to 12 -> 3 chained WMMAs per tile
  const float cb2 = sB2[lm];
  v8f e0 = {cb2, cb2, cb2, cb2, cb2, cb2, cb2, cb2};
  v8f e1 = e0;
#pragma unroll
  for (int j = 0; j < 3; ++j) {
    const int k0 = 4 * j + 2 * halfIdx;
    v2f aj0, aj1, bj;
    aj0.x = ytile[wave][0][lm * 20 + k0];
    aj0.y = ytile[wave][0][lm * 20 + k0 + 1];
    aj1.x = ytile[wave][1][lm * 20 + k0];
    aj1.y = ytile[wave][1][lm * 20 + k0 + 1];
    bj.x = sW2p[k0 * 16 + lm];          // rows 10,11 of sW2p are zero -> free K padding
    bj.y = sW2p[(k0 + 1) * 16 + lm];
    e0 = __builtin_amdgcn_wmma_f32_16x16x4_f32(false, aj0, false, bj, (short)0, e0, false, false);
    e1 = __builtin_amdgcn_wmma_f32_16x16x4_f32(false, aj1, false, bj, (short)0, e1, false, false);
  }
#pragma unroll
  for (int r = 0; r < 8; ++r) { e0[r] = fmaxf(e0[r], 0.f); e1[r] = fmaxf(e1[r], 0.f); }

  // Write Y2 back; same-wave LDS ops are in-order, loads above precede these stores.
#pragma unroll
  for (int r = 0; r < 8; ++r) {
    const int m = r + 8 * halfIdx;
    ytile[wave][0][m * 20 + lm] = e0[r];
    ytile[wave][1][m * 20 + lm] = e1[r];
  }
  __syncthreads();

  // Per-thread y[0..9]: thread t owns tile (t>>4)&1, row t&15 of its wave
  const int tl = (t >> 4) & 1, row = t & 15;
  float y[10];
#pragma unroll
  for (int u = 0; u < 10; ++u) y[u] = ytile[wave][tl][row * 20 + u];

  // ================= scalar equivariant channel =================
  // x[u,:] = lw1[u]*v everywhere => whole chain collapses to S * v
  const float nv2 = v0 * v0 + v1 * v1 + v2 * v2;
  float c1[10];
#pragma unroll
  for (int u = 0; u < 10; ++u) {
    const float s = slw1[u];
    const float n1 = fast_sqrt(nv2 * s * s + EPS2);
    c1[u] = s * FAST_TANH(n1 + snb1[u]) * fast_rcp(n1);
  }
  float S = 0.f;
#pragma unroll
  for (int vv = 0; vv < 10; ++vv) {
    float acc = 0.f;
#pragma unroll
    for (int u = 0; u < 10; ++u) acc = fmaf(c1[u], slw2[u * 10 + vv], acc);
    const float m  = acc * RSQRT_IR;
    const float n2 = fast_sqrt(m * m * nv2 + EPS2);
    const float g  = FAST_TANH(n2 + snb2[vv]) * fast_rcp(n2);
    S = fmaf(swf[vv] * y[vv], m * g, S);
  }
  S *= RSQRT_IR;
  float sx = S * v0, sy = S * v1, sz = S * v2;

  // ================= reduction over P=256 and cross product =================
#pragma unroll
  for (int off = 16; off; off >>= 1) {
    sx += __shfl_down(sx, off, 32);
    sy += __shfl_down(sy, off, 32);
    sz += __shfl_down(sz, off, 32);
  }
  if (lane == 0) { red[wave][0] = sx; red[wave][1] = sy; red[wave][2] = sz; }
  __syncthreads();
  if (t == 0) {
    float tx = 0.f, ty = 0.f, tz = 0.f;
#pragma unroll
    for (int w = 0; w < 8; ++w) { tx += red[w][0]; ty += red[w][1]; tz += red[w][2]; }
    const float ax = v1b[b * 3 + 0], ay = v1b[b * 3 + 1], az = v1b[b * 3 + 2];
    out[b * 3 + 0] = ay * tz - az * ty;
    out[b * 3 + 1] = az * tx - ax * tz;
    out[b * 3 + 2] = ax * ty - ay * tx;
  }
}

extern "C" void kernel_launch(void* const* d_in, const int* in_sizes, int n_in,
                              void* d_out, int out_size, void* d_ws, size_t ws_size,
                              hipStream_t stream) {
  (void)n_in; (void)out_size; (void)d_ws; (void)ws_size;
  const float* v1b   = (const float*)d_in[0];
  const float* invar = (const float*)d_in[1];
  const float* v2b   = (const float*)d_in[2];
  const float* W1    = (const float*)d_in[3];
  const float* b1    = (const float*)d_in[4];
  const float* W2    = (const float*)d_in[5];
  const float* b2    = (const float*)d_in[6];
  const float* lw1   = (const float*)d_in[7];
  const float* nb1   = (const float*)d_in[8];
  const float* lw2   = (const float*)d_in[9];
  const float* nb2   = (const float*)d_in[10];
  const float* wf    = (const float*)d_in[11];
  float* out = (float*)d_out;

  const int B = in_sizes[0] / 3;  // P is fixed at 256 = blockDim
  trq_2b_kernel<<<B, 256, 0, stream>>>(v1b, invar, v2b, W1, b1, W2, b2,
                                       lw1, nb1, lw2, nb2, wf, out);
}